// DTFD_MIL_19791209300142
// MI455X (gfx1250) — compile-verified
//
#include <hip/hip_runtime.h>
#include <math.h>

// ---------------------------------------------------------------------------
// DTFD-MIL forward for MI455X (gfx1250), fp32 end-to-end.
// All dense GEMMs run on V_WMMA_F32_16X16X4_F32 (wave32).
// Each wave register-blocks a 32x32 C tile (2x2 accumulators): per K-step it
// loads 2 A + 2 B fragments and issues 4 WMMAs -> 8 FLOP/byte of L2 traffic
// (2x the single-tile version). With ~150MB of tensors L2-resident (192MB L2)
// and HBM nearly idle (23.3TB/s), L2 read traffic on the 34-GFLOP channel-
// attention GEMM is the true bottleneck; this halves it. fp32 keeps the
// discrete top-k neighbor selection numerically exact.
// ---------------------------------------------------------------------------

typedef float v2f __attribute__((ext_vector_type(2)));
typedef float v8f __attribute__((ext_vector_type(8)));

__device__ __forceinline__ float sigmoidf_(float x){ return 1.0f/(1.0f+expf(-x)); }
__device__ __forceinline__ float leakyf_(float x, float s){ return x>=0.0f ? x : s*x; }

template<int ACT>
__device__ __forceinline__ float act_apply(float x){
  if (ACT==1) return fmaxf(x, 0.0f);        // relu
  if (ACT==2) return leakyf_(x, 0.01f);     // leaky 0.01
  if (ACT==3) return tanhf(x);
  if (ACT==4) return sigmoidf_(x);
  return x;
}

// C[M,N] = act(opA(A) @ opB(B) + bias)
// TRANSA=0: A row-major [M,K], access A[m*lda+k].  TRANSA=1: stored [K,M], A[k*lda+m].
// TRANSB=0: B row-major [K,N], access B[k*ldb+n].  TRANSB=1: stored [N,K], B[n*ldb+k].
// Requires: M%32==0, N%128==0, K%4==0 (all call sites satisfy this).
// 128 threads = 4 waves along N; each wave owns a 32x32 C tile (2x2 WMMA tiles).
template<int TRANSA, int TRANSB, int ACT>
__global__ void wmma_gemm(const float* __restrict__ A, const float* __restrict__ B,
                          const float* __restrict__ bias, float* __restrict__ C,
                          int M, int N, int K, int lda, int ldb, int ldc)
{
  const int lane  = threadIdx.x & 31;
  const int wv    = threadIdx.x >> 5;           // 0..3
  const int half  = lane >> 4;                  // 0: K={k,k+1}, 1: K={k+2,k+3}
  const int l16   = lane & 15;
  const int tileM = blockIdx.y * 32;
  const int tileN = blockIdx.x * 128 + wv * 32;
  if (tileM >= M || tileN >= N) return;

  v8f acc00 = {}, acc01 = {}, acc10 = {}, acc11 = {};
  const int am0 = tileM + l16;                  // A rows (M) for the two M-tiles
  const int am1 = tileM + 16 + l16;
  const int bn0 = tileN + l16;                  // B cols (N) for the two N-tiles
  const int bn1 = tileN + 16 + l16;

  for (int k = 0; k < K; k += 4) {
    const int ka = k + half * 2;
    v2f a0, a1, b0, b1;
    if (TRANSA) {
      a0.x = A[(size_t)ka*lda + am0];     a0.y = A[(size_t)(ka+1)*lda + am0];
      a1.x = A[(size_t)ka*lda + am1];     a1.y = A[(size_t)(ka+1)*lda + am1];
    } else {
      a0.x = A[(size_t)am0*lda + ka];     a0.y = A[(size_t)am0*lda + ka + 1];
      a1.x = A[(size_t)am1*lda + ka];     a1.y = A[(size_t)am1*lda + ka + 1];
    }
    if (TRANSB) {
      b0.x = B[(size_t)bn0*ldb + ka];     b0.y = B[(size_t)bn0*ldb + ka + 1];
      b1.x = B[(size_t)bn1*ldb + ka];     b1.y = B[(size_t)bn1*ldb + ka + 1];
    } else {
      b0.x = B[(size_t)ka*ldb + bn0];     b0.y = B[(size_t)(ka+1)*ldb + bn0];
      b1.x = B[(size_t)ka*ldb + bn1];     b1.y = B[(size_t)(ka+1)*ldb + bn1];
    }
    acc00 = __builtin_amdgcn_wmma_f32_16x16x4_f32(false, a0, false, b0, (short)0, acc00, false, false);
    acc01 = __builtin_amdgcn_wmma_f32_16x16x4_f32(false, a0, false, b1, (short)0, acc01, false, false);
    acc10 = __builtin_amdgcn_wmma_f32_16x16x4_f32(false, a1, false, b0, (short)0, acc10, false, false);
    acc11 = __builtin_amdgcn_wmma_f32_16x16x4_f32(false, a1, false, b1, (short)0, acc11, false, false);
  }

  // C/D layout: VGPR r, lanes 0-15 -> M = r, lanes 16-31 -> M = r+8; N = lane&15.
  #pragma unroll
  for (int r = 0; r < 8; ++r) {
    const int mr = r + half * 8;
    const int m0 = tileM + mr, m1 = tileM + 16 + mr;
    const int n0 = tileN + l16, n1 = tileN + 16 + l16;
    float v00 = acc00[r], v01 = acc01[r], v10 = acc10[r], v11 = acc11[r];
    if (bias) { v00 += bias[n0]; v01 += bias[n1]; v10 += bias[n0]; v11 += bias[n1]; }
    C[(size_t)m0*ldc + n0] = act_apply<ACT>(v00);
    C[(size_t)m0*ldc + n1] = act_apply<ACT>(v01);
    C[(size_t)m1*ldc + n0] = act_apply<ACT>(v10);
    C[(size_t)m1*ldc + n1] = act_apply<ACT>(v11);
  }
}

// ------------------------------ small kernels ------------------------------

__global__ void k_fill(float* p, float v, int n){
  int i = blockIdx.x*blockDim.x + threadIdx.x;
  if (i < n) p[i] = v;
}

// score[r] = sum_f V[r,f]*U[r,f]*w[f] + b[0]   (Fd == blockDim.x == 128)
__global__ void k_gated_score(const float* __restrict__ V, const float* __restrict__ U,
                              const float* __restrict__ w, const float* __restrict__ b,
                              float* __restrict__ out)
{
  const int r = blockIdx.x, t = threadIdx.x;
  __shared__ float red[128];
  red[t] = V[(size_t)r*128 + t] * U[(size_t)r*128 + t] * w[t];
  __syncthreads();
  for (int s = 64; s > 0; s >>= 1){ if (t < s) red[t] += red[t+s]; __syncthreads(); }
  if (t == 0) out[r] = red[0] + b[0];
}

// in-place softmax over contiguous segments of length L (one block per segment)
__global__ void k_softmax_seg(float* __restrict__ x, int L)
{
  float* p = x + (size_t)blockIdx.x * L;
  const int t = threadIdx.x;
  __shared__ float red[256];
  float m = -3.402823466e38f;
  for (int i = t; i < L; i += 256) m = fmaxf(m, p[i]);
  red[t] = m; __syncthreads();
  for (int s = 128; s > 0; s >>= 1){ if (t < s) red[t] = fmaxf(red[t], red[t+s]); __syncthreads(); }
  m = red[0]; __syncthreads();
  float sum = 0.f;
  for (int i = t; i < L; i += 256){ float e = expf(p[i] - m); p[i] = e; sum += e; }
  red[t] = sum; __syncthreads();
  for (int s = 128; s > 0; s >>= 1){ if (t < s) red[t] += red[t+s]; __syncthreads(); }
  const float inv = 1.0f / red[0];
  for (int i = t; i < L; i += 256) p[i] *= inv;
}

// af[g,f] = sum_n tmid[g*512+n, f] * w[g*512+n]   (32 groups, block = 512 threads = f)
__global__ void k_af_pool(const float* __restrict__ tmid, const float* __restrict__ w,
                          float* __restrict__ af)
{
  const int g = blockIdx.x, f = threadIdx.x;
  const float* tg = tmid + (size_t)g*512*512;
  const float* wg = w + (size_t)g*512;
  float acc = 0.f;
  for (int n = 0; n < 512; ++n) acc += tg[(size_t)n*512 + f] * wg[n];
  af[(size_t)g*512 + f] = acc;
}

// sub_preds[g*4+b, c] = af[b*8+g] . subcls_w[:,c] + b[c]  (64 threads)
__global__ void k_subpreds(const float* __restrict__ af, const float* __restrict__ w,
                           const float* __restrict__ b, float* __restrict__ out)
{
  const int idx = threadIdx.x;               // 0..63
  const int row = idx >> 1, c = idx & 1;
  const int g = row >> 2, bb = row & 3;
  const float* a = af + (size_t)(bb*8 + g) * 512;
  float acc = b[c];
  for (int f = 0; f < 512; ++f) acc += a[f] * w[f*2 + c];
  out[row*2 + c] = acc;
}

// out_mil[b,f] = sum_g att[b*8+g] * af[b*8+g, f]   (grid 4, block 512)
__global__ void k_out_mil(const float* __restrict__ af, const float* __restrict__ att,
                          float* __restrict__ om)
{
  const int b = blockIdx.x, f = threadIdx.x;
  float acc = 0.f;
  for (int g = 0; g < 8; ++g) acc += att[b*8 + g] * af[(size_t)(b*8 + g)*512 + f];
  om[(size_t)b*512 + f] = acc;
}

// generic tiny row-vs-matrix: out[r*2+c] = X[r] . w[:,c] + b[c]  (R*2 threads, K=512)
__global__ void k_rows_x_w2(const float* __restrict__ X, const float* __restrict__ w,
                            const float* __restrict__ b, float* __restrict__ out, int Kd)
{
  const int idx = threadIdx.x;
  const int r = idx >> 1, c = idx & 1;
  const float* a = X + (size_t)r * Kd;
  float acc = b[c];
  for (int f = 0; f < Kd; ++f) acc += a[f] * w[f*2 + c];
  out[r*2 + c] = acc;
}

// x_concat[r] = (r<4 ? out_mil[r] : rehearsal[r-4])   (grid 4096, block 512)
__global__ void k_concat_x(const float* __restrict__ om, const float* __restrict__ reh,
                           float* __restrict__ xc)
{
  const int r = blockIdx.x, t = threadIdx.x;
  const float* src = (r < 4) ? (om + (size_t)r*512) : (reh + (size_t)(r-4)*512);
  xc[(size_t)r*512 + t] = src[t];
}

// in-place L2 row-normalize, Fd == blockDim.x == 256
__global__ void k_rownorm(float* __restrict__ x)
{
  const int r = blockIdx.x, t = threadIdx.x;
  __shared__ float red[256];
  float v = x[(size_t)r*256 + t];
  red[t] = v*v; __syncthreads();
  for (int s = 128; s > 0; s >>= 1){ if (t < s) red[t] += red[t+s]; __syncthreads(); }
  const float inv = 1.0f / fmaxf(sqrtf(red[0]), 1e-12f);
  x[(size_t)r*256 + t] = v * inv;
}

// per-row top-8 (descending, ties -> lower index first, matching jax.lax.top_k)
__global__ void k_topk(const float* __restrict__ sim, int* __restrict__ nbr, int R, int Cn)
{
  const int r = blockIdx.x*blockDim.x + threadIdx.x;
  if (r >= R) return;
  const float* row = sim + (size_t)r * Cn;
  float bv[8]; int bi[8];
  #pragma unroll
  for (int k = 0; k < 8; ++k){ bv[k] = -3.402823466e38f; bi[k] = 0; }
  for (int j = 0; j < Cn; ++j){
    const float v = row[j];
    if (v > bv[7]){
      int k = 7;
      while (k > 0 && v > bv[k-1]){ bv[k] = bv[k-1]; bi[k] = bi[k-1]; --k; }
      bv[k] = v; bi[k] = j;
    }
  }
  #pragma unroll
  for (int k = 0; k < 8; ++k) nbr[(size_t)r*8 + k] = bi[k];
}

// edge_attr[e,f] = mean_k x_concat[nbr[e,k], f]   (grid 4096, block 512)
__global__ void k_edge_attr(const float* __restrict__ xc, const int* __restrict__ nbr,
                            float* __restrict__ ea)
{
  const int e = blockIdx.x, f = threadIdx.x;
  float acc = 0.f;
  #pragma unroll
  for (int k = 0; k < 8; ++k) acc += xc[(size_t)nbr[e*8 + k]*512 + f];
  ea[(size_t)e*512 + f] = acc * 0.125f;
}

// out[r] = X[r,:Fd] . w   (block 256)
__global__ void k_rowdot(const float* __restrict__ X, const float* __restrict__ w,
                         float* __restrict__ out, int Fd)
{
  const int r = blockIdx.x, t = threadIdx.x;
  __shared__ float red[256];
  float s = 0.f;
  for (int i = t; i < Fd; i += 256) s += X[(size_t)r*Fd + i] * w[i];
  red[t] = s; __syncthreads();
  for (int st = 128; st > 0; st >>= 1){ if (t < st) red[t] += red[t+st]; __syncthreads(); }
  if (t == 0) out[r] = red[0];
}

// monotonic-int float atomic max (pos: signed max; neg: unsigned min)
__device__ __forceinline__ void atomicMaxF(float* addr, float v){
  if (v >= 0.0f) atomicMax((int*)addr, __float_as_int(v));
  else           atomicMin((unsigned int*)addr, (unsigned int)__float_as_int(v));
}

__global__ void k_alpha_max(const int* __restrict__ nbr, const float* __restrict__ pn,
                            const float* __restrict__ pe, float* __restrict__ raw,
                            float* __restrict__ amax)
{
  const int i = blockIdx.x*blockDim.x + threadIdx.x;
  if (i >= 32768) return;
  const int n = nbr[i], e = i >> 3;
  float v = leakyf_(pn[n] + pe[e], 0.2f);
  raw[i] = v;
  atomicMaxF(&amax[n], v);
}

__global__ void k_alpha_exp(const int* __restrict__ nbr, float* __restrict__ alpha,
                            const float* __restrict__ amax, float* __restrict__ den,
                            float* __restrict__ Dv, int addDv)
{
  const int i = blockIdx.x*blockDim.x + threadIdx.x;
  if (i >= 32768) return;
  const int n = nbr[i];
  const float e = expf(alpha[i] - amax[n]);
  alpha[i] = e;
  atomicAdd(&den[n], e);
  if (addDv) atomicAdd(&Dv[n], 1.0f);
}

__global__ void k_alpha_norm(const int* __restrict__ nbr, float* __restrict__ alpha,
                             const float* __restrict__ den)
{
  const int i = blockIdx.x*blockDim.x + threadIdx.x;
  if (i >= 32768) return;
  alpha[i] = alpha[i] / den[nbr[i]];
}

// eo[e,f] = (1/8) * sum_k alpha[e*8+k] * xl[nbr[e,k], f]   (grid 4096, block 512)
__global__ void k_eo(const int* __restrict__ nbr, const float* __restrict__ alpha,
                     const float* __restrict__ xl, float* __restrict__ eo)
{
  const int e = blockIdx.x, f = threadIdx.x;
  float acc = 0.f;
  #pragma unroll
  for (int k = 0; k < 8; ++k)
    acc += alpha[e*8 + k] * xl[(size_t)nbr[e*8 + k]*512 + f];
  eo[(size_t)e*512 + f] = acc * 0.125f;
}

// conv[n,f] += (alpha[i]/Dv[n]) * eo[i>>3, f]   (grid 32768, block 512)
__global__ void k_scatter(const int* __restrict__ nbr, const float* __restrict__ alpha,
                          const float* __restrict__ Dv, const float* __restrict__ eo,
                          float* __restrict__ conv)
{
  const int i = blockIdx.x, f = threadIdx.x;
  const int n = nbr[i], e = i >> 3;
  const float coef = alpha[i] / Dv[n];        // Dv[n] >= 1 whenever node n has edges
  atomicAdd(&conv[(size_t)n*512 + f], coef * eo[(size_t)e*512 + f]);
}

// GraphNorm over 4096 rows per channel (+hgc bias, +outer leaky 0.01). block 256, grid 512.
__global__ void k_graphnorm(const float* __restrict__ src, const float* __restrict__ hb,
                            const float* __restrict__ w, const float* __restrict__ b,
                            const float* __restrict__ ms, float* __restrict__ dst)
{
  const int c = blockIdx.x, t = threadIdx.x;
  __shared__ float red[256];
  float loc[16];
  const float hbc = hb[c];
  float sum = 0.f;
  #pragma unroll
  for (int j = 0; j < 16; ++j){
    loc[j] = src[(size_t)(t + 256*j)*512 + c] + hbc;
    sum += loc[j];
  }
  red[t] = sum; __syncthreads();
  for (int st = 128; st > 0; st >>= 1){ if (t < st) red[t] += red[t+st]; __syncthreads(); }
  const float mean = red[0] * (1.0f/4096.0f);
  __syncthreads();
  const float msc = ms[c];
  float sq = 0.f;
  #pragma unroll
  for (int j = 0; j < 16; ++j){ loc[j] -= msc*mean; sq += loc[j]*loc[j]; }
  red[t] = sq; __syncthreads();
  for (int st = 128; st > 0; st >>= 1){ if (t < st) red[t] += red[t+st]; __syncthreads(); }
  const float var = red[0] * (1.0f/4096.0f);
  const float scale = w[c] * rsqrtf(var + 1e-5f);
  const float bc = b[c];
  #pragma unroll
  for (int j = 0; j < 16; ++j)
    dst[(size_t)(t + 256*j)*512 + c] = leakyf_(scale*loc[j] + bc, 0.01f);
}

// outbuf[r, 0:512|512:768|768:1024] = xc | o1 | o2   (grid 4096, block 256)
__global__ void k_concat_out(const float* __restrict__ xc, const float* __restrict__ o1,
                             const float* __restrict__ o2, float* __restrict__ ob)
{
  const int r = blockIdx.x, t = threadIdx.x;
  for (int j = t; j < 512; j += 256) ob[(size_t)r*1024 + j]       = xc[(size_t)r*512 + j];
  for (int j = t; j < 256; j += 256) ob[(size_t)r*1024 + 512 + j] = o1[(size_t)r*256 + j];
  for (int j = t; j < 256; j += 256) ob[(size_t)r*1024 + 768 + j] = o2[(size_t)r*256 + j];
}

// s[c] = sigmoid(T[c,:4096] . w2 + b2)   (grid 1024, block 256)
__global__ void k_schan(const float* __restrict__ T, const float* __restrict__ w2,
                        const float* __restrict__ b2, float* __restrict__ s)
{
  const int c = blockIdx.x, t = threadIdx.x;
  __shared__ float red[256];
  float acc = 0.f;
  for (int j = t; j < 4096; j += 256) acc += T[(size_t)c*4096 + j] * w2[j];
  red[t] = acc; __syncthreads();
  for (int st = 128; st > 0; st >>= 1){ if (t < st) red[t] += red[t+st]; __syncthreads(); }
  if (t == 0) s[c] = sigmoidf_(red[0] + b2[0]);
}

// s[c] -= mean(s)   (1 block, 1024 threads)
__global__ void k_s_center(float* __restrict__ s)
{
  const int t = threadIdx.x;
  __shared__ float red[1024];
  const float v = s[t];
  red[t] = v; __syncthreads();
  for (int st = 512; st > 0; st >>= 1){ if (t < st) red[t] += red[t+st]; __syncthreads(); }
  s[t] = v - red[0] * (1.0f/1024.0f);
}

// fusion[b,f] = sum_c ob[b,c]*s[c]*gpw[c,f] + gpb[f] + sum_k om[b,k]*mpw[k,f] + mpb[f]
__global__ void k_fusion(const float* __restrict__ ob, const float* __restrict__ s,
                         const float* __restrict__ gpw, const float* __restrict__ gpb,
                         const float* __restrict__ om, const float* __restrict__ mpw,
                         const float* __restrict__ mpb, float* __restrict__ fus)
{
  const int b = blockIdx.x, f = threadIdx.x;
  float acc = gpb[f] + mpb[f];
  for (int c = 0; c < 1024; ++c) acc += ob[(size_t)b*1024 + c] * s[c] * gpw[(size_t)c*512 + f];
  for (int k = 0; k < 512;  ++k) acc += om[(size_t)b*512 + k] * mpw[(size_t)k*512 + f];
  fus[(size_t)b*512 + f] = acc;
}

// ------------------------------ launcher -----------------------------------

extern "C" void kernel_launch(void* const* d_in, const int* in_sizes, int n_in,
                              void* d_out, int out_size, void* d_ws, size_t ws_size,
                              hipStream_t stream)
{
  (void)in_sizes; (void)n_in; (void)out_size; (void)ws_size;

  const float* x          = (const float*)d_in[0];
  const float* dimred_w   = (const float*)d_in[1];
  const float* attV_w     = (const float*)d_in[2];
  const float* attV_b     = (const float*)d_in[3];
  const float* attU_w     = (const float*)d_in[4];
  const float* attU_b     = (const float*)d_in[5];
  const float* attw_w     = (const float*)d_in[6];
  const float* attw_b     = (const float*)d_in[7];
  const float* subcls_w   = (const float*)d_in[8];
  const float* subcls_b   = (const float*)d_in[9];
  const float* lattV_w    = (const float*)d_in[10];
  const float* lattV_b    = (const float*)d_in[11];
  const float* lattU_w    = (const float*)d_in[12];
  const float* lattU_b    = (const float*)d_in[13];
  const float* lattw_w    = (const float*)d_in[14];
  const float* lattw_b    = (const float*)d_in[15];
  const float* milcls_w   = (const float*)d_in[16];
  const float* milcls_b   = (const float*)d_in[17];
  const float* rehearsal  = (const float*)d_in[18];
  const float* dsl_w      = (const float*)d_in[19];
  const float* dsl_b      = (const float*)d_in[20];
  const float* hgc1_w     = (const float*)d_in[21];
  const float* hgc1_att   = (const float*)d_in[22];
  const float* hgc1_b     = (const float*)d_in[23];
  const float* gn1_w      = (const float*)d_in[24];
  const float* gn1_b      = (const float*)d_in[25];
  const float* gn1_ms     = (const float*)d_in[26];
  const float* gfc1_w     = (const float*)d_in[27];
  const float* gfc1_b     = (const float*)d_in[28];
  const float* hgc2_w     = (const float*)d_in[29];
  const float* hgc2_att   = (const float*)d_in[30];
  const float* hgc2_b     = (const float*)d_in[31];
  const float* gn2_w      = (const float*)d_in[32];
  const float* gn2_b      = (const float*)d_in[33];
  const float* gn2_ms     = (const float*)d_in[34];
  const float* gfc2_w     = (const float*)d_in[35];
  const float* gfc2_b     = (const float*)d_in[36];
  const float* gattn1_w   = (const float*)d_in[37];
  const float* gattn1_b   = (const float*)d_in[38];
  const float* gattn2_w   = (const float*)d_in[39];
  const float* gattn2_b   = (const float*)d_in[40];
  const float* milproj_w  = (const float*)d_in[41];
  const float* milproj_b  = (const float*)d_in[42];
  const float* graphproj_w= (const float*)d_in[43];
  const float* graphproj_b= (const float*)d_in[44];
  const float* fus_w      = (const float*)d_in[45];
  const float* fus_b      = (const float*)d_in[46];

  float* Wk  = (float*)d_ws;
  float* out = (float*)d_out;   // [0..7] logits, [8..71] sub_preds, [72..79] final

  // ---- workspace layout (floats). BIG = [0, 16777216) reused per phase ----
  const size_t BIGN = 16777216ull;
  float* TMID = Wk + 0;            // 16384x512 (phase 1)
  float* HV   = Wk + 8388608;      // 16384x128
  float* HU   = Wk + 10485760;     // 16384x128
  float* SC   = Wk + 12582912;     // 16384 scores -> tAA (in-place softmax)
  float* AF   = Wk + 12599296;     // 32x512
  float* LV   = Wk + 12615680;     // 32x128
  float* LU   = Wk + 12619776;     // 32x128
  float* LA   = Wk + 12623872;     // 32
  float* SIM  = Wk + 0;            // 4096x4096 (phase 2, overlays phase-1 scratch)
  // phase-3 graph scratch inside BIG
  float* XL   = Wk + 0;            // 4096x512
  float* EAXL = Wk + 2097152;      // 4096x512
  float* PN   = Wk + 4194304;      // 4096
  float* PE   = Wk + 4198400;      // 4096
  float* ALPH = Wk + 4202496;      // 32768
  float* AMAX = Wk + 4235264;      // 4096
  float* DEN  = Wk + 4239360;      // 4096
  float* DV   = Wk + 4243456;      // 4096
  float* EO   = Wk + 4247552;      // 4096x512
  float* CONV = Wk + 6344704;      // 4096x512
  float* H2   = Wk + 8441856;      // 4096x512
  float* Tm   = Wk + 10539008;     // 1024x4096
  // persistent region beyond BIG
  float* OM   = Wk + BIGN;                 // 4x512
  float* XC   = OM  + 2048;                // 4096x512
  float* NX   = XC  + 2097152;             // 4096x256 (hx -> nx in place)
  int*   NBR  = (int*)(NX + 1048576);      // 4096x8 int
  float* EAT  = (float*)(NBR + 32768);     // 4096x512 edge_attr
  float* O1   = EAT + 2097152;             // 4096x256
  float* O2   = O1  + 1048576;             // 4096x256
  float* H    = O2  + 1048576;             // 4096x512 (conv1 result)
  float* OUTB = H   + 2097152;             // 4096x1024
  float* S    = OUTB+ 4194304;             // 1024
  float* FUS  = S   + 1024;                // 4x512

  // ======================= Phase 1: MIL attention =======================
  // tmid = relu(x @ dimred_w)        [16384,512] x [512,512]
  wmma_gemm<0,0,1><<<dim3(512/128, 16384/32), 128, 0, stream>>>(
      x, dimred_w, nullptr, TMID, 16384, 512, 512, 512, 512, 512);
  // hV = tanh(tmid@attV_w+b), hU = sigmoid(tmid@attU_w+b)  [16384,128]
  wmma_gemm<0,0,3><<<dim3(1, 16384/32), 128, 0, stream>>>(
      TMID, attV_w, attV_b, HV, 16384, 128, 512, 512, 128, 128);
  wmma_gemm<0,0,4><<<dim3(1, 16384/32), 128, 0, stream>>>(
      TMID, attU_w, attU_b, HU, 16384, 128, 512, 512, 128, 128);
  k_gated_score<<<16384, 128, 0, stream>>>(HV, HU, attw_w, attw_b, SC);
  k_softmax_seg<<<32, 256, 0, stream>>>(SC, 512);              // softmax over n per (b,g)
  k_af_pool<<<32, 512, 0, stream>>>(TMID, SC, AF);             // af[b*8+g, f]
  k_subpreds<<<1, 64, 0, stream>>>(AF, subcls_w, subcls_b, out + 8);
  // bag-level gated attention over groups (M=32)
  wmma_gemm<0,0,3><<<dim3(1, 1), 128, 0, stream>>>(AF, lattV_w, lattV_b, LV, 32, 128, 512, 512, 128, 128);
  wmma_gemm<0,0,4><<<dim3(1, 1), 128, 0, stream>>>(AF, lattU_w, lattU_b, LU, 32, 128, 512, 512, 128, 128);
  k_gated_score<<<32, 128, 0, stream>>>(LV, LU, lattw_w, lattw_b, LA);
  k_softmax_seg<<<4, 256, 0, stream>>>(LA, 8);                 // softmax over g per b
  k_out_mil<<<4, 512, 0, stream>>>(AF, LA, OM);
  k_rows_x_w2<<<1, 8, 0, stream>>>(OM, milcls_w, milcls_b, out, 512);   // logits

  // ======================= Phase 2: DSL hypergraph ======================
  k_concat_x<<<4096, 512, 0, stream>>>(OM, rehearsal, XC);
  // hx = leaky(x_concat @ dsl_w + b, 0.01); then L2 row-normalize in place
  wmma_gemm<0,0,2><<<dim3(256/128, 4096/32), 128, 0, stream>>>(
      XC, dsl_w, dsl_b, NX, 4096, 256, 512, 512, 256, 256);
  k_rownorm<<<4096, 256, 0, stream>>>(NX);
  // sim = nx @ nx^T   [4096,4096], K=256 (B accessed transposed)
  wmma_gemm<0,1,0><<<dim3(4096/128, 4096/32), 128, 0, stream>>>(
      NX, NX, nullptr, SIM, 4096, 4096, 256, 256, 256, 4096);
  k_topk<<<4096/128, 128, 0, stream>>>(SIM, NBR, 4096, 4096);
  k_edge_attr<<<4096, 512, 0, stream>>>(XC, NBR, EAT);

  // ================== Phase 3: two HypergraphConv layers ================
  auto run_conv = [&](const float* xin, const float* hw, const float* hatt,
                      const float* hb, const float* gw, const float* gb,
                      const float* gms, float* dst, int computeDv)
  {
    wmma_gemm<0,0,0><<<dim3(512/128, 4096/32), 128, 0, stream>>>(
        xin, hw, nullptr, XL, 4096, 512, 512, 512, 512, 512);
    wmma_gemm<0,0,0><<<dim3(512/128, 4096/32), 128, 0, stream>>>(
        EAT, hw, nullptr, EAXL, 4096, 512, 512, 512, 512, 512);
    k_rowdot<<<4096, 256, 0, stream>>>(XL,   hatt,       PN, 512);
    k_rowdot<<<4096, 256, 0, stream>>>(EAXL, hatt + 512, PE, 512);
    k_fill<<<16, 256, 0, stream>>>(AMAX, -3.402823466e38f, 4096);
    k_fill<<<16, 256, 0, stream>>>(DEN, 0.0f, 4096);
    if (computeDv) k_fill<<<16, 256, 0, stream>>>(DV, 0.0f, 4096);
    k_alpha_max <<<128, 256, 0, stream>>>(NBR, PN, PE, ALPH, AMAX);
    k_alpha_exp <<<128, 256, 0, stream>>>(NBR, ALPH, AMAX, DEN, DV, computeDv);
    k_alpha_norm<<<128, 256, 0, stream>>>(NBR, ALPH, DEN);
    k_eo<<<4096, 512, 0, stream>>>(NBR, ALPH, XL, EO);
    k_fill<<<(2097152 + 255)/256, 256, 0, stream>>>(CONV, 0.0f, 2097152);
    k_scatter<<<32768, 512, 0, stream>>>(NBR, ALPH, DV, EO, CONV);
    k_graphnorm<<<512, 256, 0, stream>>>(CONV, hb, gw, gb, gms, dst);
  };

  run_conv(XC, hgc1_w, hgc1_att, hgc1_b, gn1_w, gn1_b, gn1_ms, H, 1);
  wmma_gemm<0,0,2><<<dim3(256/128, 4096/32), 128, 0, stream>>>(
      H, gfc1_w, gfc1_b, O1, 4096, 256, 512, 512, 256, 256);
  run_conv(H, hgc2_w, hgc2_att, hgc2_b, gn2_w, gn2_b, gn2_ms, H2, 0);
  wmma_gemm<0,0,2><<<dim3(256/128, 4096/32), 128, 0, stream>>>(
      H2, gfc2_w, gfc2_b, O2, 4096, 256, 512, 512, 256, 256);

  // =============== channel attention + fusion + final head ==============
  k_concat_out<<<4096, 256, 0, stream>>>(XC, O1, O2, OUTB);
  // T = relu(out^T @ gattn1_w + b)   M=1024, K=4096, N=4096 (A transposed)
  wmma_gemm<1,0,1><<<dim3(4096/128, 1024/32), 128, 0, stream>>>(
      OUTB, gattn1_w, gattn1_b, Tm, 1024, 4096, 4096, 1024, 4096, 4096);
  k_schan<<<1024, 256, 0, stream>>>(Tm, gattn2_w, gattn2_b, S);
  k_s_center<<<1, 1024, 0, stream>>>(S);
  k_fusion<<<4, 512, 0, stream>>>(OUTB, S, graphproj_w, graphproj_b,
                                  OM, milproj_w, milproj_b, FUS);
  k_rows_x_w2<<<1, 8, 0, stream>>>(FUS, fus_w, fus_b, out + 72, 512);
}